// MultiViewFPNet_73761768341678
// MI455X (gfx1250) — compile-verified
//
#include <hip/hip_runtime.h>

// MI455X / gfx1250. Output-store-bound kernel (170 MB f32 out, ~7.3 us floor @ 23.3 TB/s).
// Per-(b,v) batched congruence transform vec(R^T M R) = (R^T (x) R^T) vec(M) done with
// V_WMMA_F32_16X16X4_F32 (full f32 precision): out[32x9] = Mmat[32x9] * T[9x9],
// rows 0..15 = vec(A_l), rows 16..31 = vec(Cm_l), padded K 9->12 (3 chunks of K=4).

typedef float v2f __attribute__((ext_vector_type(2)));
typedef float v8f __attribute__((ext_vector_type(8)));

#define NLIMB 16
#define NJ    17
#define DIM   51           // 3*NJ
#define TILE  (DIM * DIM)  // 2601

__global__ __launch_bounds__(256) void fpnet_block_tridiag_kernel(
    const float* __restrict__ pts,   // (B*V, NLIMB, 3)
    const float* __restrict__ nvs,   // (B*V, NLIMB, 3)
    const float* __restrict__ Rs,    // (B*V, 3, 3) row-major
    const float* __restrict__ w,     // (B*V, NLIMB)
    float* __restrict__ out)         // (B*V, 51, 51)
{
    __shared__ float lds_M[32][12];  // rows 0..15: vec(A_l), rows 16..31: vec(Cm_l) (k = 3a+b)
    __shared__ float lds_w[NLIMB];
    __shared__ float lds_SA[NLIMB][9];  //  w_l * vec(R^T A_l  R)  (n = 3i+j)
    __shared__ float lds_SC[NLIMB][9];  // -w_l * vec(R^T Cm_l R)

    const int bv  = blockIdx.x;
    const int tid = threadIdx.x;

    // ---- Phase 1: per-limb 3x3 math (closed form). A = Ns - q q^T/denom, Cm = q q^T/denom,
    //      with n = normalize(nv), q = p - n (n.p), denom = 2 (p.q).
    if (tid < NLIMB) {
        const int base = (bv * NLIMB + tid) * 3;
        const float p0 = pts[base + 0], p1 = pts[base + 1], p2 = pts[base + 2];
        const float m0 = nvs[base + 0], m1 = nvs[base + 1], m2 = nvs[base + 2];
        const float rn = rsqrtf(m0 * m0 + m1 * m1 + m2 * m2);
        const float n0 = m0 * rn, n1 = m1 * rn, n2 = m2 * rn;
        const float ndp = n0 * p0 + n1 * p1 + n2 * p2;
        const float q0 = p0 - n0 * ndp, q1 = p1 - n1 * ndp, q2 = p2 - n2 * ndp;
        const float inv = 1.0f / (2.0f * (p0 * q0 + p1 * q1 + p2 * q2));
        const float nn[3] = {n0, n1, n2};
        const float qq[3] = {q0, q1, q2};
#pragma unroll
        for (int a = 0; a < 3; ++a) {
#pragma unroll
            for (int b = 0; b < 3; ++b) {
                const float g = qq[a] * qq[b] * inv;
                lds_M[tid][3 * a + b]      = (a == b ? 1.0f : 0.0f) - nn[a] * nn[b] - g; // A
                lds_M[16 + tid][3 * a + b] = g;                                          // Cm
            }
        }
        lds_w[tid] = w[bv * NLIMB + tid];
    }
    __syncthreads();

    // ---- Phase 2+3: wave 0 runs the batched sandwich as 6x V_WMMA_F32_16X16X4_F32.
    // threads 0..31 == wave 0 (wave32), EXEC all ones inside this branch.
    if (tid < 32) {
        float r[9];
#pragma unroll
        for (int k = 0; k < 9; ++k) r[k] = Rs[bv * 9 + k];

        const int half = tid >> 4;   // 0: lanes 0-15, 1: lanes 16-31
        const int l16  = tid & 15;

        // B operand (T[9x9], K padded to 12): VGPR0 holds K = 4c+2*half, VGPR1 K = 4c+2*half+1.
        // T[k][n] = R[a][i]*R[b][j], k=3a+b, n=3i+j.
        v2f Bc[3];
#pragma unroll
        for (int c = 0; c < 3; ++c) {
            const int k0 = 4 * c + 2 * half;
            float b0 = 0.0f, b1 = 0.0f;
            if (l16 < 9) {
                const int i = l16 / 3, j = l16 % 3;
                if (k0 < 9)     b0 = r[3 * (k0 / 3) + i] * r[3 * (k0 % 3) + j];
                if (k0 + 1 < 9) b1 = r[3 * ((k0 + 1) / 3) + i] * r[3 * ((k0 + 1) % 3) + j];
            }
            Bc[c].x = b0;
            Bc[c].y = b1;
        }

#pragma unroll
        for (int t = 0; t < 2; ++t) {  // t=0: A matrices, t=1: Cm matrices
            v8f acc = {0.f, 0.f, 0.f, 0.f, 0.f, 0.f, 0.f, 0.f};
#pragma unroll
            for (int c = 0; c < 3; ++c) {
                const int k0 = 4 * c + 2 * half;
                v2f Ac;  // 16x4 A layout: lane holds row M=l16; VGPR0 K=k0, VGPR1 K=k0+1
                Ac.x = (k0 < 9)     ? lds_M[16 * t + l16][k0]     : 0.0f;
                Ac.y = (k0 + 1 < 9) ? lds_M[16 * t + l16][k0 + 1] : 0.0f;
                acc = __builtin_amdgcn_wmma_f32_16x16x4_f32(
                    /*neg_a=*/false, Ac, /*neg_b=*/false, Bc[c],
                    /*c_mod=*/(short)0, acc, /*reuse_a=*/false, /*reuse_b=*/false);
            }
            // C/D layout: VGPR rr -> M = rr + 8*half, N = l16.
            if (l16 < 9) {
#pragma unroll
                for (int rr = 0; rr < 8; ++rr) {
                    const int m = rr + 8 * half;      // limb index within tile
                    const float s = lds_w[m];
                    if (t == 0) lds_SA[m][l16] = s * acc[rr];
                    else        lds_SC[m][l16] = -s * acc[rr];
                }
            }
        }
    }
    __syncthreads();

    // ---- Phase 4: stream the 51x51 tile (block-tridiagonal band; chain pairs (l, l+1)).
    float* o = out + (size_t)bv * TILE;
    for (int idx = tid; idx < TILE; idx += 256) {
        const int row = idx / DIM, col = idx - row * DIM;
        const int jr = row / 3, i = row - 3 * jr;
        const int jc = col / 3, j = col - 3 * jc;
        const int n = 3 * i + j;
        float val = 0.0f;
        if (jr == jc) {                 // diag block d gets w_d*S_d and w_{d-1}*S_{d-1}
            if (jr < NLIMB) val += lds_SA[jr][n];
            if (jr >= 1)    val += lds_SA[jr - 1][n];
        } else if (jc == jr + 1) {      // upper off-diag: limb jr (jr <= 15 here)
            val = lds_SC[jr][n];
        } else if (jr == jc + 1) {      // lower off-diag: limb jc
            val = lds_SC[jc][n];
        }
        o[idx] = val;
    }
}

extern "C" void kernel_launch(void* const* d_in, const int* in_sizes, int n_in,
                              void* d_out, int out_size, void* d_ws, size_t ws_size,
                              hipStream_t stream) {
    (void)n_in; (void)out_size; (void)d_ws; (void)ws_size;
    const float* pts = (const float*)d_in[0];
    const float* nvs = (const float*)d_in[1];
    const float* Rs  = (const float*)d_in[2];
    const float* w   = (const float*)d_in[3];
    // d_in[4] = limb_pairs (chain (l, l+1) by construction), d_in[5] = n_joints (=17)
    float* out = (float*)d_out;

    const int nbv = in_sizes[3] / NLIMB;  // B*V = 16384
    fpnet_block_tridiag_kernel<<<nbv, 256, 0, stream>>>(pts, nvs, Rs, w, out);
}